// DGCNNv2Encoder_35098472743109
// MI455X (gfx1250) — compile-verified
//
#include <hip/hip_runtime.h>
#include <hip/hip_bf16.h>

typedef __attribute__((ext_vector_type(2))) float v2f;
typedef __attribute__((ext_vector_type(8))) float v8f;

#define BATCH 8
#define NPTS  2048
#define KNN   20
#define BN    (BATCH * NPTS)
#define EPSV  1e-5f

static __device__ __forceinline__ v8f wmma_f32x4(v2f a, v2f b, v8f c) {
  // V_WMMA_F32_16X16X4_F32 : D = A(16x4) * B(4x16) + C(16x16)
  return __builtin_amdgcn_wmma_f32_16x16x4_f32(
      /*neg_a=*/false, a, /*neg_b=*/false, b,
      /*c_mod=*/(short)0, c, /*reuse_a=*/false, /*reuse_b=*/false);
}

// ---------------------------------------------------------------------------
// Pad (BN,3) points into (BN,4) with zero 4th channel: removes all bounds
// guards from the layer-1 WMMA paths.
// ---------------------------------------------------------------------------
__global__ void pad_points(const float* __restrict__ p, float* __restrict__ xp) {
  int i = blockIdx.x * blockDim.x + threadIdx.x;
  if (i >= BN) return;
  xp[i * 4 + 0] = p[i * 3 + 0];
  xp[i * 4 + 1] = p[i * 3 + 1];
  xp[i * 4 + 2] = p[i * 3 + 2];
  xp[i * 4 + 3] = 0.f;
}

// ---------------------------------------------------------------------------
// Squared norms per point row.
// ---------------------------------------------------------------------------
__global__ void norms_k(const float* __restrict__ X, int ldx, int Cdim,
                        float* __restrict__ out, int total) {
  int i = blockIdx.x * blockDim.x + threadIdx.x;
  if (i >= total) return;
  const float* r = X + (size_t)i * ldx;
  float s = 0.f;
  for (int c = 0; c < Cdim; ++c) s += r[c] * r[c];
  out[i] = s;
}

// ---------------------------------------------------------------------------
// Fused KNN: WMMA Gram-matrix tiles + register-resident top-20 selection.
// Block = 512 threads (16 waves); block owns a 16-row stripe of one batch.
// ---------------------------------------------------------------------------
template <int CDIM>
__global__ void __launch_bounds__(512)
knn_topk(const float* __restrict__ X, int ldx,
         const float* __restrict__ norms, int* __restrict__ idxOut) {
  __shared__ float Atile[16][CDIM];
  __shared__ float Chunk[16][256];

  const int tilesPerBatch = NPTS / 16;
  const int b = blockIdx.x / tilesPerBatch;
  const int rowBase = (blockIdx.x % tilesPerBatch) * 16;
  const float* Xb = X + (size_t)b * NPTS * ldx;
  const float* nb = norms + (size_t)b * NPTS;

  const int tid = threadIdx.x;
  const int wave = tid >> 5;
  const int lane = tid & 31;
  const int half = lane >> 4;
  const int l16 = lane & 15;

  for (int i = tid; i < 16 * CDIM; i += 512) {
    int r = i / CDIM, c = i % CDIM;  // CDIM compile-time -> shifts
    Atile[r][c] = Xb[(size_t)(rowBase + r) * ldx + c];
  }
  __syncthreads();

  float tv[KNN];
  int ti[KNN];
#pragma unroll
  for (int i = 0; i < KNN; ++i) { tv[i] = -__builtin_inff(); ti[i] = 0x7fffffff; }

  for (int chunk = 0; chunk < NPTS; chunk += 256) {
    const int colBase = chunk + wave * 16;
    v8f acc = {0.f, 0.f, 0.f, 0.f, 0.f, 0.f, 0.f, 0.f};
    const float* colRow = Xb + (size_t)(colBase + l16) * ldx;
#pragma unroll 16
    for (int k0 = 0; k0 < CDIM; k0 += 4) {
      int ka = k0 + 2 * half;
      v2f a = *(const v2f*)&Atile[l16][ka];   // ds_load_b64
      v2f bb = *(const v2f*)&colRow[ka];      // global_load_b64
      acc = wmma_f32x4(a, bb, acc);
    }
    const float cn = nb[colBase + l16];
#pragma unroll
    for (int v = 0; v < 8; ++v) {
      int r = v + 8 * half;
      Chunk[r][wave * 16 + l16] = 2.f * acc[v] - cn;  // row-order preserving
    }
    __syncthreads();

    // Selection: wave handles row `wave`; lane scans 8 strided columns.
#pragma unroll
    for (int s = 0; s < 8; ++s) {
      int lc = lane + 32 * s;
      float val = Chunk[wave][lc];
      int col = chunk + lc;
      if (val > tv[KNN - 1]) {
        float cv = val; int ci = col;
#pragma unroll
        for (int p = 0; p < KNN; ++p) {  // branchless sorted (desc) insert
          if (cv > tv[p]) {
            float t0 = tv[p]; tv[p] = cv; cv = t0;
            int t1 = ti[p]; ti[p] = ci; ci = t1;
          }
        }
      }
    }
    __syncthreads();
  }

  // Merge 32 per-lane lists: 20 rounds of wave arg-max over list heads.
  int* orow = idxOut + ((size_t)b * NPTS + rowBase + wave) * KNN;
  for (int r = 0; r < KNN; ++r) {
    float bv = tv[0]; int bi = ti[0];
#pragma unroll
    for (int off = 16; off > 0; off >>= 1) {
      float ov = __shfl_xor(bv, off, 32);
      int oi = __shfl_xor(bi, off, 32);
      if (ov > bv || (ov == bv && oi < bi)) { bv = ov; bi = oi; }
    }
    if (ti[0] == bi) {  // winning lane pops its head
#pragma unroll
      for (int p = 0; p < KNN - 1; ++p) { tv[p] = tv[p + 1]; ti[p] = ti[p + 1]; }
      tv[KNN - 1] = -__builtin_inff(); ti[KNN - 1] = 0x7fffffff;
    }
    if (lane == 0) orow[r] = bi;
  }
}

// ---------------------------------------------------------------------------
// Combined edge weights, zero-padded to Cp columns:
//   rows [0,O)  = W[:, :C]            (applies to x_j)
//   rows [O,2O) = W[:, C:] - W[:, :C] (applies to x_i)
// ---------------------------------------------------------------------------
__global__ void prep_w(const float* __restrict__ W, int O, int C, int Cp,
                       float* __restrict__ Wc) {
  int i = blockIdx.x * blockDim.x + threadIdx.x;
  if (i >= 2 * O * Cp) return;
  int o = i / Cp, c = i % Cp;
  float v = 0.f;
  if (c < C)
    v = (o < O) ? W[o * 2 * C + c]
                : W[(o - O) * 2 * C + C + c] - W[(o - O) * 2 * C + c];
  Wc[i] = v;
}

// ---------------------------------------------------------------------------
// WMMA GEMM: Out[M x Nout] = X[M x KDIM] @ Wr^T,  Wr is (Nout x KDIM).
// Block = 256 threads (8 waves); block tile = 16 rows x 128 cols.
// ---------------------------------------------------------------------------
template <int KDIM>
__global__ void __launch_bounds__(256)
gemm16(const float* __restrict__ X, int ldx, const float* __restrict__ Wr,
       float* __restrict__ Out, int ldo, int Nout) {
  __shared__ float Atile[16][KDIM];
  const int tid = threadIdx.x;
  const int wave = tid >> 5, lane = tid & 31;
  const int half = lane >> 4, l16 = lane & 15;
  const int rowBase = blockIdx.x * 16;

  for (int i = tid; i < 16 * KDIM; i += 256) {
    int r = i / KDIM, c = i % KDIM;
    Atile[r][c] = X[(size_t)(rowBase + r) * ldx + c];
  }
  __syncthreads();

  const int colBase = blockIdx.y * 128 + wave * 16;
  if (colBase >= Nout) return;

  v8f acc = {0.f, 0.f, 0.f, 0.f, 0.f, 0.f, 0.f, 0.f};
  const float* wrow = Wr + (size_t)(colBase + l16) * KDIM;
#pragma unroll 16
  for (int k0 = 0; k0 < KDIM; k0 += 4) {
    int ka = k0 + 2 * half;
    v2f a = *(const v2f*)&Atile[l16][ka];
    v2f bb = *(const v2f*)&wrow[ka];
    acc = wmma_f32x4(a, bb, acc);
  }
#pragma unroll
  for (int v = 0; v < 8; ++v) {
    int r = v + 8 * half;
    Out[(size_t)(rowBase + r) * ldo + colBase + l16] = acc[v];
  }
}

// ---------------------------------------------------------------------------
// EdgeConv epilogue: out[n,o] = max_j lrelu(s*(z[m_j,o] + t[n,o]) + beta)
// Z rows hold [z(O) | t(O)]; block = O threads, one block per (b,n).
// ---------------------------------------------------------------------------
__global__ void gather_max(const float* __restrict__ Z, int ldz, int O,
                           const int* __restrict__ Idx,
                           const float* __restrict__ gamma,
                           const float* __restrict__ beta,
                           float* __restrict__ Out, int ldo) {
  const int bn = blockIdx.x;
  const int o = threadIdx.x;
  const int base = (bn / NPTS) * NPTS;
  const float s = gamma[o] * rsqrtf(1.f + EPSV);
  const float bt = beta[o];
  const float t = Z[(size_t)bn * ldz + O + o];
  const int* ip = Idx + (size_t)bn * KNN;
  float m = -__builtin_inff();
#pragma unroll 4
  for (int j = 0; j < KNN; ++j) {
    int mrow = ip[j];
    float v = Z[(size_t)(base + mrow) * ldz + o] + t;
    float y = s * v + bt;
    y = (y >= 0.f) ? y : 0.2f * y;
    m = fmaxf(m, y);
  }
  Out[(size_t)bn * ldo + o] = m;
}

// ---------------------------------------------------------------------------
// Final layer: bn_lrelu(xcat @ W5^T) fused with max/mean pooling (atomics).
// K = 512 fixed; Nout = 1024. Block tile 16 rows x 128 cols.
// ---------------------------------------------------------------------------
__global__ void __launch_bounds__(256)
gemm_pool(const float* __restrict__ X, const float* __restrict__ W5,
          const float* __restrict__ gamma, const float* __restrict__ beta,
          float* __restrict__ maxacc, float* __restrict__ sumacc) {
  __shared__ float Atile[16][512];
  const int tid = threadIdx.x;
  const int wave = tid >> 5, lane = tid & 31;
  const int half = lane >> 4, l16 = lane & 15;
  const int rowBase = blockIdx.x * 16;
  const int b = rowBase / NPTS;

  for (int i = tid; i < 16 * 512; i += 256) {
    int r = i >> 9, c = i & 511;
    Atile[r][c] = X[(size_t)(rowBase + r) * 512 + c];
  }
  __syncthreads();

  const int colBase = blockIdx.y * 128 + wave * 16;
  v8f acc = {0.f, 0.f, 0.f, 0.f, 0.f, 0.f, 0.f, 0.f};
  const float* wrow = W5 + (size_t)(colBase + l16) * 512;
#pragma unroll 16
  for (int k0 = 0; k0 < 512; k0 += 4) {
    int ka = k0 + 2 * half;
    v2f a = *(const v2f*)&Atile[l16][ka];
    v2f bb = *(const v2f*)&wrow[ka];
    acc = wmma_f32x4(a, bb, acc);
  }

  const int oc = colBase + l16;
  const float s = gamma[oc] * rsqrtf(1.f + EPSV);
  const float bt = beta[oc];
  float lmax = -__builtin_inff(), lsum = 0.f;
#pragma unroll
  for (int v = 0; v < 8; ++v) {
    float y = s * acc[v] + bt;
    y = (y >= 0.f) ? y : 0.2f * y;
    lmax = fmaxf(lmax, y);
    lsum += y;
  }
  lmax = fmaxf(lmax, __shfl_xor(lmax, 16, 32));
  lsum += __shfl_xor(lsum, 16, 32);
  if (half == 0) {
    atomicMax(&maxacc[b * 1024 + oc], lmax);
    atomicAdd(&sumacc[b * 1024 + oc], lsum);
  }
}

__global__ void init_acc(float* __restrict__ mx, float* __restrict__ sm, int n) {
  int i = blockIdx.x * blockDim.x + threadIdx.x;
  if (i < n) { mx[i] = -__builtin_inff(); sm[i] = 0.f; }
}

__global__ void finalize_k(const float* __restrict__ mx,
                           const float* __restrict__ sm,
                           float* __restrict__ out) {
  int i = blockIdx.x * blockDim.x + threadIdx.x;
  if (i >= BATCH * 1024) return;
  int b = i >> 10, o = i & 1023;
  out[b * 2048 + o] = mx[i];
  out[b * 2048 + 1024 + o] = sm[i] * (1.f / NPTS);
}

// ---------------------------------------------------------------------------
template <int CDIM>
static void run_edge_layer(const float* xin, int ldx, int Wsrc_C, int O, int off,
                           const float* Wp, const float* gm, const float* bb,
                           float* xcat, float* zbuf, int* idxb, float* norms,
                           float* wcomb, hipStream_t stream) {
  const int O2 = 2 * O;
  norms_k<<<(BN + 255) / 256, 256, 0, stream>>>(xin, ldx, CDIM, norms, BN);
  knn_topk<CDIM><<<BN / 16, 512, 0, stream>>>(xin, ldx, norms, idxb);
  prep_w<<<(O2 * CDIM + 255) / 256, 256, 0, stream>>>(Wp, O, Wsrc_C, CDIM, wcomb);
  gemm16<CDIM><<<dim3(BN / 16, (O2 + 127) / 128), 256, 0, stream>>>(
      xin, ldx, wcomb, zbuf, O2, O2);
  gather_max<<<BN, O, 0, stream>>>(zbuf, O2, O, idxb, gm, bb, xcat + off, 512);
}

extern "C" void kernel_launch(void* const* d_in, const int* in_sizes, int n_in,
                              void* d_out, int out_size, void* d_ws, size_t ws_size,
                              hipStream_t stream) {
  (void)in_sizes; (void)n_in; (void)out_size; (void)ws_size;
  const float* pts = (const float*)d_in[0];
  const float* W[5] = {(const float*)d_in[1], (const float*)d_in[2],
                       (const float*)d_in[3], (const float*)d_in[4],
                       (const float*)d_in[5]};
  const float* g[5] = {(const float*)d_in[6], (const float*)d_in[8],
                       (const float*)d_in[10], (const float*)d_in[12],
                       (const float*)d_in[14]};
  const float* bt[5] = {(const float*)d_in[7], (const float*)d_in[9],
                        (const float*)d_in[11], (const float*)d_in[13],
                        (const float*)d_in[15]};

  float* xcat = (float*)d_ws;                  // (BN, 512)
  float* zbuf = xcat + (size_t)BN * 512;       // (BN, up to 512)
  int* idxb = (int*)(zbuf + (size_t)BN * 512); // (BN, 20)
  float* norms = (float*)(idxb + (size_t)BN * KNN);
  float* wcomb = norms + BN;                   // up to 2*256*128 floats
  float* maxacc = wcomb + 131072;              // (B, 1024)
  float* sumacc = maxacc + BATCH * 1024;       // (B, 1024)
  float* xpad = sumacc + BATCH * 1024;         // (BN, 4)

  pad_points<<<(BN + 255) / 256, 256, 0, stream>>>(pts, xpad);

  run_edge_layer<4>(xpad, 4, 3, 64, 0, W[0], g[0], bt[0],
                    xcat, zbuf, idxb, norms, wcomb, stream);
  run_edge_layer<64>(xcat + 0, 512, 64, 64, 64, W[1], g[1], bt[1],
                     xcat, zbuf, idxb, norms, wcomb, stream);
  run_edge_layer<64>(xcat + 64, 512, 64, 128, 128, W[2], g[2], bt[2],
                     xcat, zbuf, idxb, norms, wcomb, stream);
  run_edge_layer<128>(xcat + 128, 512, 128, 256, 256, W[3], g[3], bt[3],
                      xcat, zbuf, idxb, norms, wcomb, stream);

  init_acc<<<(BATCH * 1024 + 255) / 256, 256, 0, stream>>>(maxacc, sumacc,
                                                           BATCH * 1024);
  gemm_pool<<<dim3(BN / 16, 8), 256, 0, stream>>>(xcat, W[4], g[4], bt[4],
                                                  maxacc, sumacc);
  finalize_k<<<(BATCH * 1024 + 255) / 256, 256, 0, stream>>>(maxacc, sumacc,
                                                             (float*)d_out);
}